// LinearEncoder_46093589021378
// MI455X (gfx1250) — compile-verified
//
#include <hip/hip_runtime.h>
#include <hip/hip_bf16.h>

#define N_NODES 50000
#define N_EDGES 1600000
#define IN_C    128
#define OUT_C   64

typedef __attribute__((ext_vector_type(2))) float v2f;
typedef __attribute__((ext_vector_type(8))) float v8f;

// ---------------------------------------------------------------------------
// K0: deg[i] = 1.0  (self-loop contribution)
// ---------------------------------------------------------------------------
__global__ void gcn_init_deg(float* __restrict__ deg) {
    int i = blockIdx.x * blockDim.x + threadIdx.x;
    if (i < N_NODES) deg[i] = 1.0f;
}

// ---------------------------------------------------------------------------
// K1: deg[col[e]] += 1 for every edge (unit edge weights)
// ---------------------------------------------------------------------------
__global__ void gcn_degree(const int* __restrict__ col, float* __restrict__ deg) {
    int e = blockIdx.x * blockDim.x + threadIdx.x;
    if (e < N_EDGES) atomicAdd(&deg[col[e]], 1.0f);
}

// ---------------------------------------------------------------------------
// K2: dinv[i] = rsqrt(deg[i])   (deg >= 1 always, no zero case)
// ---------------------------------------------------------------------------
__global__ void gcn_dinv(float* __restrict__ deg) {
    int i = blockIdx.x * blockDim.x + threadIdx.x;
    if (i < N_NODES) deg[i] = rsqrtf(deg[i]);
}

// ---------------------------------------------------------------------------
// K3: h = x @ W via V_WMMA_F32_16X16X4_F32, full fp32 precision.
// One wave32 per 16x64 output row-block: the A fragment (16 rows of x) is
// loaded ONCE per k-step and fed to 4 WMMAs (one per 16-col N-tile) with 4
// independent accumulators -> x traffic 25.6MB (not 4x), and 4 independent
// accumulation chains pipeline back-to-back on the matrix unit.
//
// A (16x4 f32) layout:  lane = {l16 = M, half = lane>>4};
//   VGPR0 = A[M][K=2*half], VGPR1 = A[M][K=2*half+1]
// B (4x16 f32) layout (transpose-symmetric):
//   VGPR0 = B[K=2*half][N=l16], VGPR1 = B[K=2*half+1][N=l16]
// C/D (16x16 f32): VGPR r, lanes 0-15 -> M=r, lanes 16-31 -> M=r+8; N=l16.
// ---------------------------------------------------------------------------
__global__ void gcn_gemm_wmma(const float* __restrict__ x,
                              const float* __restrict__ W,
                              float* __restrict__ h) {
    const int mt    = blockIdx.x;        // 0 .. 3124
    const int lane  = threadIdx.x;       // 0..31 (wave32)
    const int l16   = lane & 15;
    const int half  = lane >> 4;

    const int mbase = mt * 16;

    const float* __restrict__ xrow = x + (size_t)(mbase + l16) * IN_C;

    v8f c0 = {}, c1 = {}, c2 = {}, c3 = {};
#pragma unroll
    for (int k = 0; k < IN_C; k += 4) {
        // A fragment: contiguous pair -> single b64 load, shared by 4 WMMAs
        v2f a;
        a.x = xrow[k + 2 * half];
        a.y = xrow[k + 2 * half + 1];

        const float* __restrict__ w0 = W + (size_t)(k + 2 * half) * OUT_C + l16;
        const float* __restrict__ w1 = w0 + OUT_C;

        v2f b0, b1, b2, b3;
        b0.x = w0[0];  b0.y = w1[0];
        b1.x = w0[16]; b1.y = w1[16];
        b2.x = w0[32]; b2.y = w1[32];
        b3.x = w0[48]; b3.y = w1[48];

        // (neg_a, A, neg_b, B, c_mod, C, reuse_a, reuse_b)
        c0 = __builtin_amdgcn_wmma_f32_16x16x4_f32(false, a, false, b0, (short)0, c0, false, false);
        c1 = __builtin_amdgcn_wmma_f32_16x16x4_f32(false, a, false, b1, (short)0, c1, false, false);
        c2 = __builtin_amdgcn_wmma_f32_16x16x4_f32(false, a, false, b2, (short)0, c2, false, false);
        c3 = __builtin_amdgcn_wmma_f32_16x16x4_f32(false, a, false, b3, (short)0, c3, false, false);
    }

#pragma unroll
    for (int r = 0; r < 8; ++r) {
        const size_t row = (size_t)(mbase + r + 8 * half);
        float* __restrict__ hp = h + row * OUT_C + l16;
        hp[0]  = c0[r];
        hp[16] = c1[r];
        hp[32] = c2[r];
        hp[48] = c3[r];
    }
}

// ---------------------------------------------------------------------------
// K4: out = h * dinv^2 + bias  (self-loop term; also initializes d_out)
// One thread per float4 (16 threads per node).
// ---------------------------------------------------------------------------
__global__ void gcn_selfloop_bias(const float* __restrict__ h,
                                  const float* __restrict__ dinv,
                                  const float* __restrict__ bias,
                                  float* __restrict__ out) {
    int t = blockIdx.x * blockDim.x + threadIdx.x;
    if (t >= N_NODES * (OUT_C / 4)) return;
    int i  = t >> 4;            // node
    int c4 = (t & 15) * 4;      // channel group
    float d = dinv[i];
    float s = d * d;
    float4 hv = *(const float4*)(h + (size_t)i * OUT_C + c4);
    float4 bv = *(const float4*)(bias + c4);
    float4 o;
    o.x = hv.x * s + bv.x;
    o.y = hv.y * s + bv.y;
    o.z = hv.z * s + bv.z;
    o.w = hv.w * s + bv.w;
    *(float4*)(out + (size_t)i * OUT_C + c4) = o;
}

// ---------------------------------------------------------------------------
// K5: per-edge gather / scale / scatter-add.
// 16 threads per edge, each owns a float4 slice -> b128 gather (L2-resident h)
// + 4 global_atomic_add_f32 (no return -> STOREcnt path).
// ---------------------------------------------------------------------------
__global__ void gcn_scatter(const int* __restrict__ row,
                            const int* __restrict__ col,
                            const float* __restrict__ h,
                            const float* __restrict__ dinv,
                            float* __restrict__ out) {
    int t = blockIdx.x * blockDim.x + threadIdx.x;   // E*16 = 25.6M < 2^31
    if (t >= N_EDGES * (OUT_C / 4)) return;
    int e  = t >> 4;
    int c4 = (t & 15) * 4;
    int s  = row[e];
    int d  = col[e];
    float norm = dinv[s] * dinv[d];
    float4 hv = *(const float4*)(h + (size_t)s * OUT_C + c4);
    float* op = out + (size_t)d * OUT_C + c4;
    atomicAdd(op + 0, hv.x * norm);
    atomicAdd(op + 1, hv.y * norm);
    atomicAdd(op + 2, hv.z * norm);
    atomicAdd(op + 3, hv.w * norm);
}

// ---------------------------------------------------------------------------
extern "C" void kernel_launch(void* const* d_in, const int* in_sizes, int n_in,
                              void* d_out, int out_size, void* d_ws, size_t ws_size,
                              hipStream_t stream) {
    const float* x    = (const float*)d_in[0];       // [N, 128]
    const int*   ei   = (const int*)d_in[1];         // [2, E] row-major
    const float* W    = (const float*)d_in[2];       // [128, 64]
    const float* bias = (const float*)d_in[3];       // [64]
    float*       out  = (float*)d_out;               // [N, 64]

    // workspace: deg/dinv [N] floats, then h [N, 64] floats (16B aligned:
    // N*4 = 200000 bytes is a multiple of 16, hipMalloc base is 256-aligned)
    float* deg = (float*)d_ws;
    float* h   = deg + N_NODES;

    const int* src = ei;             // edge_index[0]
    const int* dst = ei + N_EDGES;   // edge_index[1]

    gcn_init_deg<<<(N_NODES + 255) / 256, 256, 0, stream>>>(deg);
    gcn_degree  <<<(N_EDGES + 255) / 256, 256, 0, stream>>>(dst, deg);
    gcn_dinv    <<<(N_NODES + 255) / 256, 256, 0, stream>>>(deg);

    gcn_gemm_wmma<<<N_NODES / 16, 32, 0, stream>>>(x, W, h);

    gcn_selfloop_bias<<<(N_NODES * (OUT_C / 4) + 255) / 256, 256, 0, stream>>>(
        h, deg, bias, out);

    gcn_scatter<<<(N_EDGES * (OUT_C / 4) + 255) / 256, 256, 0, stream>>>(
        src, dst, h, deg, out);
}